// NeighborhoodAggregation_8392366096430
// MI455X (gfx1250) — compile-verified
//
#include <hip/hip_runtime.h>
#include <hip/hip_bf16.h>

typedef unsigned int uint32x4 __attribute__((ext_vector_type(4)));
typedef int          int32x8  __attribute__((ext_vector_type(8)));
typedef int          int32x4  __attribute__((ext_vector_type(4)));

#define TILE_EDGES 512   // divides 1,600,000 exactly; tail handled anyway
#define D_FEAT     64

// ---------------------------------------------------------------------------
// Kernel 1: out_real = Z_real, out_imag = Z_imag (self loop), counts = 1.0
// ---------------------------------------------------------------------------
__global__ void nag_init(const float4* __restrict__ zr, const float4* __restrict__ zi,
                         float4* __restrict__ ar, float4* __restrict__ ai,
                         float* __restrict__ counts, int n4, int n_nodes) {
    int i = blockIdx.x * blockDim.x + threadIdx.x;
    if (i < n4) {
        ar[i] = zr[i];
        ai[i] = zi[i];
    }
    if (i < n_nodes) counts[i] = 1.0f;
}

// ---------------------------------------------------------------------------
// Kernel 2: per-edge scatter-add. Edge indices staged into LDS via the
// Tensor Data Mover (one 2D descriptor: 512 src + 512 dst indices), then
// 32 lanes cooperate per edge: lane owns columns 2*lane, 2*lane+1 of both
// the real and imag rows -> coalesced float2 gathers (L2-resident) and
// hardware global_atomic_add_f32 scatters.
// ---------------------------------------------------------------------------
__global__ void nag_edges(const float2* __restrict__ zr2, const float2* __restrict__ zi2,
                          const int* __restrict__ edges,   // flat (2, n_edges): src row 0, dst row 1
                          float* __restrict__ ar, float* __restrict__ ai,
                          float* __restrict__ counts, int n_edges) {
    __shared__ int tile[2 * TILE_EDGES];   // [0..511]=src, [512..1023]=dst (LDS offset 0)

    const int e0 = blockIdx.x * TILE_EDGES;

#if __has_builtin(__builtin_amdgcn_tensor_load_to_lds)
    if (threadIdx.x < 32) {   // wave 0 issues the TDM descriptor
        unsigned long long ga = (unsigned long long)(const void*)(edges + e0); // tile start (src row)
        int rem = n_edges - e0;
        if (rem > TILE_EDGES) rem = TILE_EDGES;

        // D# group 0: count=1 valid, lds_addr=0, 57-bit global addr, type=2 ("image")
        uint32x4 g0;
        g0.x = 1u;
        g0.y = 0u;                                            // lds_addr (bytes)
        g0.z = (unsigned int)ga;
        g0.w = (unsigned int)((ga >> 32) & 0x01FFFFFFull) | 0x80000000u; // addr[56:32] | type=2

        // D# group 1 (bit-packed per ISA 8.4):
        //  w0: mask=0, data_size=2 (4B), no barrier/iterate/pad
        //  w1[31:16] = tensor_dim0[15:0]
        //  w2[15:0]  = tensor_dim0[31:16], w2[31:16] = tensor_dim1 (=2 rows)
        //  w3[31:16] = tile_dim0 (=512)
        //  w4[15:0]  = tile_dim1 (=2), tile_dim2=0
        //  w5        = tensor_dim0_stride[31:0] (= n_edges)
        unsigned int td0 = (unsigned int)rem;
        int32x8 g1;
        g1[0] = (int)(2u << 16);
        g1[1] = (int)((td0 & 0xFFFFu) << 16);
        g1[2] = (int)((td0 >> 16) | (2u << 16));
        g1[3] = (int)(((unsigned int)TILE_EDGES) << 16);
        g1[4] = (int)2u;
        g1[5] = (int)(unsigned int)n_edges;
        g1[6] = 0;
        g1[7] = 0;

        int32x4 z4 = {0, 0, 0, 0};                 // groups 2/3 unused for 2D tile (tile_dim2=0)
        int32x8 z8 = {0, 0, 0, 0, 0, 0, 0, 0};     // extra operand in 6-arg toolchain form
        __builtin_amdgcn_tensor_load_to_lds(g0, g1, z4, z4, z8, 0);
        __builtin_amdgcn_s_wait_tensorcnt(0);
    }
#else
    // Fallback: plain coalesced staging loads
    for (int i = threadIdx.x; i < 2 * TILE_EDGES; i += blockDim.x) {
        int col = (i < TILE_EDGES) ? i : (i - TILE_EDGES);
        int row = (i < TILE_EDGES) ? 0 : 1;
        int g = e0 + col;
        tile[i] = (g < n_edges) ? edges[(size_t)row * (size_t)n_edges + (size_t)g] : 0;
    }
#endif

    // The TDM writes LDS behind the compiler's back; escape the tile pointer
    // through an asm memory clobber so the subsequent reads are real ds_loads
    // and not folded to poison (tile is never written by visible code).
    {
        int* tp = &tile[0];
        asm volatile("" : "+v"(tp) : : "memory");
    }
    __syncthreads();

    const int lane = threadIdx.x & 31;
    const int wave = threadIdx.x >> 5;   // 8 waves per block

    for (int e = wave; e < TILE_EDGES; e += 8) {
        if (e0 + e >= n_edges) break;
        const int s = tile[e];
        const int d = tile[TILE_EDGES + e];

        // gather: lane owns columns 2*lane, 2*lane+1 (coalesced 256B/wave/array)
        const float2 vr = zr2[(size_t)s * (D_FEAT / 2) + lane];
        const float2 vi = zi2[(size_t)s * (D_FEAT / 2) + lane];

        float* pr = ar + (size_t)d * D_FEAT + 2 * lane;
        float* pi = ai + (size_t)d * D_FEAT + 2 * lane;
        unsafeAtomicAdd(pr,     vr.x);
        unsafeAtomicAdd(pr + 1, vr.y);
        unsafeAtomicAdd(pi,     vi.x);
        unsafeAtomicAdd(pi + 1, vi.y);
        if (lane == 0) unsafeAtomicAdd(counts + d, 1.0f);
    }
}

// ---------------------------------------------------------------------------
// Kernel 3: out /= max(counts, 1)
// ---------------------------------------------------------------------------
__global__ void nag_norm(float4* __restrict__ ar, float4* __restrict__ ai,
                         const float* __restrict__ counts, int n4) {
    int i = blockIdx.x * blockDim.x + threadIdx.x;
    if (i >= n4) return;
    int node = i >> 4;   // 16 float4 per 64-float row
    float c = counts[node];
    float sc = 1.0f / fmaxf(c, 1.0f);
    float4 r = ar[i];
    r.x *= sc; r.y *= sc; r.z *= sc; r.w *= sc;
    ar[i] = r;
    float4 m = ai[i];
    m.x *= sc; m.y *= sc; m.z *= sc; m.w *= sc;
    ai[i] = m;
}

extern "C" void kernel_launch(void* const* d_in, const int* in_sizes, int n_in,
                              void* d_out, int out_size, void* d_ws, size_t ws_size,
                              hipStream_t stream) {
    const float* Zr   = (const float*)d_in[0];
    const float* Zi   = (const float*)d_in[1];
    const int*  edges = (const int*)d_in[2];   // flat (2, n_edges) int32

    const int n_nodes = in_sizes[0] / D_FEAT;
    const int n_edges = in_sizes[2] / 2;

    float* Ar     = (float*)d_out;
    float* Ai     = (float*)d_out + (size_t)n_nodes * D_FEAT;
    float* counts = (float*)d_ws;              // n_nodes floats of scratch

    const int n4 = n_nodes * (D_FEAT / 4);     // float4 elements per array

    nag_init<<<(n4 + 255) / 256, 256, 0, stream>>>(
        (const float4*)Zr, (const float4*)Zi, (float4*)Ar, (float4*)Ai,
        counts, n4, n_nodes);

    const int nblocks = (n_edges + TILE_EDGES - 1) / TILE_EDGES;
    nag_edges<<<nblocks, 256, 0, stream>>>(
        (const float2*)Zr, (const float2*)Zi, edges, Ar, Ai, counts, n_edges);

    nag_norm<<<(n4 + 255) / 256, 256, 0, stream>>>(
        (float4*)Ar, (float4*)Ai, counts, n4);
}